// SRUCell_25718264168738
// MI455X (gfx1250) — compile-verified
//
#include <hip/hip_runtime.h>
#include <math.h>

typedef __attribute__((ext_vector_type(2))) float v2f;
typedef __attribute__((ext_vector_type(8))) float v8f;

#define L_SEQ 1024
#define BATCH 32
#define DIM   512
#define N_TOT (3 * DIM)          // 1536
#define M_TOT (L_SEQ * BATCH)    // 32768

// GEMM tiling
#define BLK_M 128                // rows per block (8 waves x 16)
#define BLK_N 64                 // cols per block (4 wmma tiles per wave)
#define BLK_K 32                 // k-chunk staged in LDS
#define LDA   36                 // padded LDS stride for A tile (floats), %4==0, conflict-free
#define LDB2  160                // dwords per B kpair-row: 64 cols * 2 + 32 pad (stride % 64 == 32)

// ---------------------------------------------------------------------------
// Kernel 1: u = x @ W  using V_WMMA_F32_16X16X4_F32 (fp32 matrix pipe)
// ---------------------------------------------------------------------------
__global__ __launch_bounds__(256)
void sru_gemm_u(const float* __restrict__ X, const float* __restrict__ W,
                float* __restrict__ U)
{
    __shared__ float As[BLK_M * LDA];           // 128 x 32 (+pad)  = 18.0 KB
    __shared__ float Bs[(BLK_K / 2) * LDB2];    // 16 kpair x 64 x2 = 10.0 KB

    const int tid  = threadIdx.x;
    const int lane = tid & 31;
    const int wave = tid >> 5;          // 0..7
    const int m    = lane & 15;         // row-in-tile (A/D) / col-in-tile (B/D)
    const int hh   = lane >> 4;         // lane-half selector (K pair / M+8)

    const int blockRow = blockIdx.x * BLK_M;
    const int colBase  = blockIdx.y * BLK_N;

    v8f acc0 = {}, acc1 = {}, acc2 = {}, acc3 = {};

    for (int k0 = 0; k0 < DIM; k0 += BLK_K) {
        // ---- stage A tile (128 rows x 32 k), coalesced float4 loads ----
        {
            int row = tid >> 1;                 // 0..127, 2 threads per row
            int seg = (tid & 1) * 16;           // 16 floats each
            const float4* src =
                (const float4*)(X + (size_t)(blockRow + row) * DIM + k0 + seg);
            float4 p0 = src[0], p1 = src[1], p2 = src[2], p3 = src[3];
            float* dst = &As[row * LDA + seg];
            *(float4*)(dst + 0)  = p0;
            *(float4*)(dst + 4)  = p1;
            *(float4*)(dst + 8)  = p2;
            *(float4*)(dst + 12) = p3;
        }
        // ---- stage B tile (32 k x 64 cols) in K-pair interleaved layout ----
        // Bs[kp*LDB2 + 2*col] = { W[k0+2kp][col], W[k0+2kp+1][col] }
        {
            int kp = tid >> 4;                  // 0..15 kpair row
            int c  = (tid & 15) * 4;            // 4 cols per thread
            const float4 r0 =
                *(const float4*)(W + (size_t)(k0 + 2 * kp)     * N_TOT + colBase + c);
            const float4 r1 =
                *(const float4*)(W + (size_t)(k0 + 2 * kp + 1) * N_TOT + colBase + c);
            float* dst = &Bs[kp * LDB2 + c * 2];
            v2f t0; t0.x = r0.x; t0.y = r1.x;
            v2f t1; t1.x = r0.y; t1.y = r1.y;
            v2f t2; t2.x = r0.z; t2.y = r1.z;
            v2f t3; t3.x = r0.w; t3.y = r1.w;
            ((v2f*)dst)[0] = t0;
            ((v2f*)dst)[1] = t1;
            ((v2f*)dst)[2] = t2;
            ((v2f*)dst)[3] = t3;
        }
        __syncthreads();

        // A fragment source row for this wave's 16x64 strip
        const float* arow = &As[(wave * 16 + m) * LDA];

        #pragma unroll
        for (int kk = 0; kk < BLK_K; kk += 4) {
            // A 16x4 fp32 fragment: lanes 0-15 hold K={kk,kk+1}, lanes 16-31 K={kk+2,kk+3}
            v2f a = *(const v2f*)(arow + kk + 2 * hh);
            // B fragment: one b64 load of the interleaved K-pair, per N tile
            const float* bp = &Bs[((kk >> 1) + hh) * LDB2 + m * 2];
            v2f b0 = *(const v2f*)(bp + 0);
            acc0 = __builtin_amdgcn_wmma_f32_16x16x4_f32(false, a, false, b0,
                                                         (short)0, acc0, false, false);
            v2f b1 = *(const v2f*)(bp + 32);
            acc1 = __builtin_amdgcn_wmma_f32_16x16x4_f32(false, a, false, b1,
                                                         (short)0, acc1, false, false);
            v2f b2 = *(const v2f*)(bp + 64);
            acc2 = __builtin_amdgcn_wmma_f32_16x16x4_f32(false, a, false, b2,
                                                         (short)0, acc2, false, false);
            v2f b3 = *(const v2f*)(bp + 96);
            acc3 = __builtin_amdgcn_wmma_f32_16x16x4_f32(false, a, false, b3,
                                                         (short)0, acc3, false, false);
        }
        __syncthreads();
    }

    // ---- write D: VGPR v -> M = v + 8*hh, N = m ----
    const int rowBase = blockRow + wave * 16 + 8 * hh;
    float* o0 = U + (size_t)rowBase * N_TOT + colBase + m;
    #pragma unroll
    for (int v = 0; v < 8; ++v) {
        float* o = o0 + (size_t)v * N_TOT;
        o[0]  = acc0[v];
        o[16] = acc1[v];
        o[32] = acc2[v];
        o[48] = acc3[v];
    }
}

// ---------------------------------------------------------------------------
// Kernel 2: sequential SRU scan over L; one thread per (b, j) state lane
// ---------------------------------------------------------------------------
__global__ __launch_bounds__(256)
void sru_scan(const float* __restrict__ U, const float* __restrict__ X,
              const float* __restrict__ WC, const float* __restrict__ BIAS,
              const float* __restrict__ C0,
              float* __restrict__ H, float* __restrict__ CLAST)
{
    const int idx = blockIdx.x * blockDim.x + threadIdx.x;  // 0..16383 = b*512 + j
    const int j = idx & (DIM - 1);
    const int b = idx >> 9;

    const float vf = WC[j];
    const float vr = WC[DIM + j];
    const float bf = BIAS[j];
    const float br = BIAS[DIM + j];
    float c = C0[idx];

    const float SCALE_X = 1.7320508075688772f;   // sqrt(3)

    size_t uoff = (size_t)b * N_TOT + j;   // u[t, b, gate, j] = U[(t*B+b)*1536 + gate*512 + j]
    size_t xoff = (size_t)idx;             // x[t, b, j]

    for (int t = 0; t < L_SEQ; ++t) {
        float u0 = U[uoff];
        float u1 = U[uoff + DIM];
        float u2 = U[uoff + 2 * DIM];
        float xt = X[xoff];

        float zf = u1 + vf * c + bf;
        float f  = 1.0f / (1.0f + expf(-zf));
        c = f * c + (1.0f - f) * u0;

        float zr = u2 + vr * c + br;
        float r  = 1.0f / (1.0f + expf(-zr));
        float g  = tanhf(c);

        H[xoff] = r * g + (1.0f - r) * (xt * SCALE_X);

        uoff += (size_t)BATCH * N_TOT;   // +49152
        xoff += (size_t)BATCH * DIM;     // +16384
    }
    CLAST[idx] = c;
}

// ---------------------------------------------------------------------------
extern "C" void kernel_launch(void* const* d_in, const int* in_sizes, int n_in,
                              void* d_out, int out_size, void* d_ws, size_t ws_size,
                              hipStream_t stream)
{
    const float* x    = (const float*)d_in[0];   // (L, B, d)
    const float* w    = (const float*)d_in[1];   // (d, 3d)
    const float* wc   = (const float*)d_in[2];   // (2d,)
    const float* bias = (const float*)d_in[3];   // (2d,)
    const float* c0   = (const float*)d_in[4];   // (B, d)

    float* h     = (float*)d_out;                          // (L, B, d)
    float* clast = h + (size_t)L_SEQ * BATCH * DIM;        // (B, d)
    float* u     = (float*)d_ws;                           // (L*B, 3d) scratch, 192 MiB

    dim3 gblk(M_TOT / BLK_M, N_TOT / BLK_N);   // (256, 24)
    sru_gemm_u<<<gblk, 256, 0, stream>>>(x, w, u);

    sru_scan<<<(BATCH * DIM) / 256, 256, 0, stream>>>(u, x, wc, bias, c0, h, clast);
}